// SSM_1468878815316
// MI455X (gfx1250) — compile-verified
//
#include <hip/hip_runtime.h>
#include <math.h>

// S4-DPLR SSM forward for MI455X (gfx1250, wave32).
// Pipeline (all f32 / complex-f32, memory-bound => keep traffic coalesced):
//   K1 at_roots[l,div] = c*(Cc2 @ diag(r) @ B)   -- WMMA f32 16x16x4 complex GEMM
//   T1 transpose [l][q] -> [q][l]                 (coalesced FFT-over-l reads)
//   K3 per (div,i,j): ifft2048 -> pad -> fft4096  (LDS Stockham)
//   T2 transpose [q][bin] -> [bin][q]             (coalesced einsum reads)
//   K2 Ud = fft4096(pad(u))                       (LDS Stockham)
//   K4 Yf[b,l,i] = sum_j Kd[l,i,j]*Ud[b,l,j]     -- WMMA f32 16x16x4 complex GEMM
//   K5 y = real(ifft4096(hermitian(Yf)))[:2048]
// Requires ws_size >= 1088 MiB (two 544 MiB ping-pong regions).

#define LSEQ   2048
#define NFFT   4096
#define NBINS  2049
#define NDIV   8
#define NBATCH 8
#define DIMTOT 512
#define QTOT   32768   // NDIV*64*64
#define PI_F   3.14159265358979323846f

typedef __attribute__((ext_vector_type(2))) float v2f;
typedef __attribute__((ext_vector_type(8))) float v8f;

// D = A(16x4,f32) * B(4x16,f32) + C(16x16,f32); NEG/C-mod unused.
#define WMMA_F32_16x16x4(A, B, C) \
  __builtin_amdgcn_wmma_f32_16x16x4_f32(false, (A), false, (B), (short)0, (C), false, false)

__device__ __forceinline__ float2 cmk(float x, float y) { return make_float2(x, y); }
__device__ __forceinline__ float2 cadd(float2 a, float2 b) { return cmk(a.x + b.x, a.y + b.y); }
__device__ __forceinline__ float2 csub(float2 a, float2 b) { return cmk(a.x - b.x, a.y - b.y); }
__device__ __forceinline__ float2 cmul(float2 a, float2 b) {
  return cmk(a.x * b.x - a.y * b.y, a.x * b.y + a.y * b.x);
}
__device__ __forceinline__ float2 cconj(float2 a) { return cmk(a.x, -a.y); }
__device__ __forceinline__ float2 cscale(float s, float2 a) { return cmk(s * a.x, s * a.y); }
__device__ __forceinline__ float2 cdiv(float2 a, float2 b) {
  float inv = 1.0f / (b.x * b.x + b.y * b.y);
  return cmk((a.x * b.x + a.y * b.y) * inv, (a.y * b.x - a.x * b.y) * inv);
}

// Self-sorting Stockham radix-2, LDS ping-pong. sign=-1 forward, +1 inverse
// (unnormalized). Returns buffer holding the result. Call with full block.
__device__ float2* fft_stockham(float2* b0, float2* b1, int n, float sign) {
  float2* src = b0;
  float2* dst = b1;
  for (int half = n >> 1, m = 1; half >= 1; half >>= 1, m <<= 1) {
    __syncthreads();
    for (int t = threadIdx.x; t < (n >> 1); t += blockDim.x) {
      int j = t / m;
      int k = t - j * m;
      float2 a = src[k + m * j];
      float2 b = src[k + m * (j + half)];
      float sn, cs;
      __sincosf(sign * PI_F * (float)j / (float)half, &sn, &cs);
      dst[k + 2 * m * j]     = cadd(a, b);
      dst[k + 2 * m * j + m] = cmul(cmk(cs, sn), csub(a, b));
    }
    float2* tmp = src; src = dst; dst = tmp;
  }
  __syncthreads();
  return src;
}

// ---------------- K1: at_roots via complex WMMA GEMM -------------------------
// block = (l, div), 256 threads (8 waves). LDS: sA (Cc2*c) 32KB + sB (r*B) 32KB.
__global__ __launch_bounds__(256) void k_atroots(
    const float* __restrict__ Lre, const float* __restrict__ Lim,
    const float* __restrict__ Pre, const float* __restrict__ Pim,
    const float* __restrict__ Bre, const float* __restrict__ Bim,
    const float* __restrict__ Cre, const float* __restrict__ Cim,
    const float* __restrict__ stepp, float2* __restrict__ atroots) {
  __shared__ float2 sA[64][64];
  __shared__ float2 sB[64][64];
  const int l = blockIdx.x, dv = blockIdx.y, tid = threadIdx.x;

  const float s = __expf(stepp[0]);
  float sn, cs;
  __sincosf(-2.0f * PI_F * (float)l / (float)LSEQ, &sn, &cs);
  const float2 om  = cmk(cs, sn);
  const float2 den = cadd(cmk(1.f, 0.f), om);
  const float2 g   = cscale(2.0f / s, cdiv(csub(cmk(1.f, 0.f), om), den));
  const float2 c2  = cdiv(cmk(2.f, 0.f), den);

  if (tid < 64) {  // r[n] -> sA[0][n]
    float lre = fminf(Lre[tid * NDIV + dv], -1e-4f);
    sA[0][tid] = cdiv(cmk(1.f, 0.f), csub(g, cmk(lre, Lim[tid * NDIV + dv])));
  }
  __syncthreads();
  if (tid < 64) {  // k01[i] -> sA[1][i]
    float2 acc = cmk(0.f, 0.f);
    for (int n = 0; n < 64; ++n) {
      float2 cc = cmk(Cre[(tid * 64 + n) * NDIV + dv], -Cim[(tid * 64 + n) * NDIV + dv]);
      float2 p  = cmk(Pre[n * NDIV + dv], Pim[n * NDIV + dv]);
      acc = cadd(acc, cmul(cc, cmul(sA[0][n], p)));
    }
    sA[1][tid] = acc;
  } else if (tid == 64) {  // s11 = 1/(1+k11) -> sA[2][0]   (pc*p = |p|^2 real)
    float2 k11 = cmk(0.f, 0.f);
    for (int n = 0; n < 64; ++n) {
      float2 p = cmk(Pre[n * NDIV + dv], Pim[n * NDIV + dv]);
      k11 = cadd(k11, cscale(p.x * p.x + p.y * p.y, sA[0][n]));
    }
    sA[2][0] = cdiv(cmk(1.f, 0.f), cadd(cmk(1.f, 0.f), k11));
  }
  __syncthreads();
  float2 myk01 = cmk(0.f, 0.f);
  const float2 s11 = sA[2][0];
  if (tid < 64) myk01 = sA[1][tid];
  for (int idx = tid; idx < 4096; idx += 256) {  // sB[n][j] = r[n]*B[n][j]
    int n = idx >> 6, j = idx & 63;
    float2 b = cmk(Bre[(n * 64 + j) * NDIV + dv], Bim[(n * 64 + j) * NDIV + dv]);
    sB[n][j] = cmul(sA[0][n], b);
  }
  __syncthreads();
  if (tid < 64) {  // sA[i][n] = c * (Cc[i][n] - s11*k01[i]*pc[n])
    float2 sk = cmul(s11, myk01);
    for (int n = 0; n < 64; ++n) {
      float2 cc = cmk(Cre[(tid * 64 + n) * NDIV + dv], -Cim[(tid * 64 + n) * NDIV + dv]);
      float2 pc = cmk(Pre[n * NDIV + dv], -Pim[n * NDIV + dv]);
      sA[tid][n] = cmul(c2, csub(cc, cmul(sk, pc)));
    }
  }
  __syncthreads();

  // Complex 64x64x64 GEMM: 16 output tiles of 16x16, 2 per wave, K in steps of 4.
  const int lane = tid & 31, wv = tid >> 5;
  const int mr = lane & 15, hi = lane >> 4;
  const size_t base = (size_t)(l * NDIV + dv) * 4096;
  for (int tt = 0; tt < 2; ++tt) {
    const int tile = wv * 2 + tt;
    const int m0 = (tile >> 2) * 16, n0 = (tile & 3) * 16;
    v8f dre = {0, 0, 0, 0, 0, 0, 0, 0};
    v8f dim = {0, 0, 0, 0, 0, 0, 0, 0};
    for (int k0 = 0; k0 < 64; k0 += 4) {
      float2 a0 = sA[m0 + mr][k0 + 2 * hi];
      float2 a1 = sA[m0 + mr][k0 + 2 * hi + 1];
      float2 b0 = sB[k0 + 2 * hi][n0 + mr];
      float2 b1 = sB[k0 + 2 * hi + 1][n0 + mr];
      v2f ar = {a0.x, a1.x}, ai = {a0.y, a1.y}, an = {-a0.y, -a1.y};
      v2f br = {b0.x, b1.x}, bi = {b0.y, b1.y};
      dre = WMMA_F32_16x16x4(ar, br, dre);
      dre = WMMA_F32_16x16x4(an, bi, dre);
      dim = WMMA_F32_16x16x4(ar, bi, dim);
      dim = WMMA_F32_16x16x4(ai, br, dim);
    }
    for (int q = 0; q < 8; ++q) {
      int row = m0 + q + hi * 8, col = n0 + mr;
      atroots[base + row * 64 + col] = cmk(dre[q], dim[q]);
    }
  }
}

// ------------- tiled complex transpose: in[R][C] -> out[C][R] ----------------
__global__ __launch_bounds__(256) void k_transpose(
    const float2* __restrict__ in, float2* __restrict__ out, int R, int C) {
  __shared__ float2 tile[32][33];
  const int c0 = blockIdx.x * 32, r0 = blockIdx.y * 32, x = threadIdx.x;
  for (int yy = threadIdx.y; yy < 32; yy += 8) {
    int r = r0 + yy, c = c0 + x;
    if (r < R && c < C) tile[yy][x] = in[(size_t)r * C + c];
  }
  __syncthreads();
  for (int yy = threadIdx.y; yy < 32; yy += 8) {
    int r = r0 + x, c = c0 + yy;
    if (r < R && c < C) out[(size_t)c * R + r] = tile[x][yy];
  }
}

// ---- K3: per (div,i,j): K[t]=real(ifft2048(at)); Kd=fft4096(pad(K)) ---------
__global__ __launch_bounds__(256) void k_kernel_fft(
    const float2* __restrict__ atT, float2* __restrict__ KdT) {
  __shared__ float2 sf[2][NFFT];
  const size_t row = blockIdx.x;
  for (int t = threadIdx.x; t < LSEQ; t += blockDim.x)
    sf[0][t] = atT[row * LSEQ + t];
  float2* res = fft_stockham(sf[0], sf[1], LSEQ, +1.f);
  for (int t = threadIdx.x; t < NFFT; t += blockDim.x) {
    float2 v = (t < LSEQ) ? cmk(res[t].x * (1.f / LSEQ), 0.f) : cmk(0.f, 0.f);
    sf[0][t] = v;  // per-thread same-index read->write; fft's barrier guards rest
  }
  float2* res2 = fft_stockham(sf[0], sf[1], NFFT, -1.f);
  for (int t = threadIdx.x; t < NBINS; t += blockDim.x)
    KdT[row * NBINS + t] = res2[t];
}

// ---------------- K2: Ud[b][bin][ch] = fft4096(pad(u[b,:,ch])) ---------------
__global__ __launch_bounds__(256) void k_fft_u(
    const float* __restrict__ u, float2* __restrict__ Ud) {
  __shared__ float2 sf[2][NFFT];
  const int b = blockIdx.x, ch = blockIdx.y;
  for (int t = threadIdx.x; t < NFFT; t += blockDim.x)
    sf[0][t] = (t < LSEQ) ? cmk(u[((size_t)b * LSEQ + t) * DIMTOT + ch], 0.f)
                          : cmk(0.f, 0.f);
  float2* res = fft_stockham(sf[0], sf[1], NFFT, -1.f);
  for (int t = threadIdx.x; t < NBINS; t += blockDim.x)
    Ud[((size_t)b * NBINS + t) * DIMTOT + ch] = res[t];
}

// --- K4: per (bin,div): Yf[b,i] = sum_j Kd[i,j]*Ud[b,j] via complex WMMA -----
// 128 threads = 4 waves, wave w owns output columns i in [16w,16w+16).
// A = Ud (M=16, rows 8..15 zero), B[k=j][n=i] = Kd[bin][div][i][j].
__global__ __launch_bounds__(128) void k_freq_mm(
    const float2* __restrict__ Kd, const float2* __restrict__ Ud,
    float2* __restrict__ Yf) {
  const int l = blockIdx.x, dv = blockIdx.y;
  const int lane = threadIdx.x & 31, wv = threadIdx.x >> 5;
  const int nr = lane & 15, hi = lane >> 4;
  const int i0 = wv * 16;
  const size_t kbase = (size_t)l * QTOT + (size_t)dv * 4096;
  v8f dre = {0, 0, 0, 0, 0, 0, 0, 0};
  v8f dim = {0, 0, 0, 0, 0, 0, 0, 0};
  for (int k0 = 0; k0 < 64; k0 += 4) {
    const int ka = k0 + 2 * hi;
    float2 a0 = cmk(0.f, 0.f), a1 = cmk(0.f, 0.f);
    if (nr < 8) {  // A row = batch index
      a0 = Ud[((size_t)nr * NBINS + l) * DIMTOT + (ka + 0) * NDIV + dv];
      a1 = Ud[((size_t)nr * NBINS + l) * DIMTOT + (ka + 1) * NDIV + dv];
    }
    float2 b0 = Kd[kbase + (size_t)(i0 + nr) * 64 + (ka + 0)];
    float2 b1 = Kd[kbase + (size_t)(i0 + nr) * 64 + (ka + 1)];
    v2f ar = {a0.x, a1.x}, ai = {a0.y, a1.y}, an = {-a0.y, -a1.y};
    v2f br = {b0.x, b1.x}, bi = {b0.y, b1.y};
    dre = WMMA_F32_16x16x4(ar, br, dre);
    dre = WMMA_F32_16x16x4(an, bi, dre);
    dim = WMMA_F32_16x16x4(ar, bi, dim);
    dim = WMMA_F32_16x16x4(ai, br, dim);
  }
  if (hi == 0) {  // D rows 0..7 (lanes 0-15) are the valid batch rows
    for (int q = 0; q < 8; ++q)
      Yf[((size_t)q * NBINS + l) * DIMTOT + (i0 + nr) * NDIV + dv] =
          cmk(dre[q], dim[q]);
  }
}

// ------- K5: y[b,:,ch] = real(ifft4096(hermitian_extend(Yf)))[:2048] ---------
__global__ __launch_bounds__(256) void k_ifft_y(
    const float2* __restrict__ Yf, float* __restrict__ out) {
  __shared__ float2 sf[2][NFFT];
  const int b = blockIdx.x, ch = blockIdx.y;
  for (int t = threadIdx.x; t < NFFT; t += blockDim.x) {
    float2 v = (t < NBINS)
                   ? Yf[((size_t)b * NBINS + t) * DIMTOT + ch]
                   : cconj(Yf[((size_t)b * NBINS + (NFFT - t)) * DIMTOT + ch]);
    sf[0][t] = v;
  }
  float2* res = fft_stockham(sf[0], sf[1], NFFT, +1.f);
  for (int t = threadIdx.x; t < LSEQ; t += blockDim.x)
    out[((size_t)b * LSEQ + t) * DIMTOT + ch] = res[t].x * (1.f / NFFT);
}

extern "C" void kernel_launch(void* const* d_in, const int* in_sizes, int n_in,
                              void* d_out, int out_size, void* d_ws,
                              size_t ws_size, hipStream_t stream) {
  (void)in_sizes; (void)n_in; (void)out_size; (void)ws_size;
  const float* u    = (const float*)d_in[0];
  const float* Lre  = (const float*)d_in[1];
  const float* Lim  = (const float*)d_in[2];
  const float* Pre  = (const float*)d_in[3];
  const float* Pim  = (const float*)d_in[4];
  const float* Bre  = (const float*)d_in[5];
  const float* Bim  = (const float*)d_in[6];
  const float* Cre  = (const float*)d_in[7];
  const float* Cim  = (const float*)d_in[8];
  const float* stp  = (const float*)d_in[9];

  // Two 544 MiB ping-pong regions; buffers reuse dead space (stream-ordered).
  const size_t REG = 544ull << 20;
  float2* at  = (float2*)d_ws;                        // [2048][32768]
  float2* atT = (float2*)((char*)d_ws + REG);         // [32768][2048]
  float2* KdT = (float2*)d_ws;                        // [32768][2049] (over at)
  float2* Kd  = (float2*)((char*)d_ws + REG);         // [2049][32768] (over atT)
  float2* Ud  = (float2*)d_ws;                        // [8][2049][512] (over KdT)
  float2* Yf  = (float2*)((char*)d_ws + (128ull << 20));  // [8][2049][512]

  k_atroots<<<dim3(LSEQ, NDIV), 256, 0, stream>>>(Lre, Lim, Pre, Pim, Bre, Bim,
                                                  Cre, Cim, stp, at);
  k_transpose<<<dim3(QTOT / 32, LSEQ / 32), dim3(32, 8), 0, stream>>>(
      at, atT, LSEQ, QTOT);
  k_kernel_fft<<<QTOT, 256, 0, stream>>>(atT, KdT);
  k_transpose<<<dim3((NBINS + 31) / 32, QTOT / 32), dim3(32, 8), 0, stream>>>(
      KdT, Kd, QTOT, NBINS);
  k_fft_u<<<dim3(NBATCH, DIMTOT), 256, 0, stream>>>(u, Ud);
  k_freq_mm<<<dim3(NBINS, NDIV), 128, 0, stream>>>(Kd, Ud, Yf);
  k_ifft_y<<<dim3(NBATCH, DIMTOT), 256, 0, stream>>>(Yf, (float*)d_out);
}